// Person2Relation_36833639531229
// MI455X (gfx1250) — compile-verified
//
#include <hip/hip_runtime.h>

// ---------------------------------------------------------------------------
// Person2Relation on MI455X (gfx1250, wave32, WMMA + async LDS copy engine)
//
//   h   = relu(concat(f_i, f_j) @ W1^T + b1)  -> factored per-person partials
//   out = h @ W2^T + b2
//
// Kernel 1: Aws/Bws = person_feats @ W1-halves^T   (2048x512x512 GEMM, x2)
// Kernel 2: gather+bias+relu into LDS, then 63488x512x512 GEMM vs W2
// Matrix core: V_WMMA_F32_16X16X4_F32 (fp32, matches reference math)
// Data movement: GLOBAL_LOAD_ASYNC_TO_LDS_B128, double-buffered W chunks
// ---------------------------------------------------------------------------

typedef __attribute__((ext_vector_type(2))) float v2f;
typedef __attribute__((ext_vector_type(8))) float v8f;

#define D_  512
#define NP  32
#define BB  64
#define PP  992            // 32*31 ordered pairs
#define M1  2048           // BB*NP rows for layer-1 partials
#define M2  63488          // BB*PP pair rows
#define TM  64             // M-tile per block
#define KC  32             // K chunk staged in LDS (double buffered)
#define NCHUNK (D_ / KC)   // 16
#define SA_STRIDE 516      // 512 + 4 dword pad (16 distinct banks across frag lanes)
#define SW_STRIDE 36       // 32 + 4 dword pad (36n mod 64 distinct for n=0..15)
#define SA_DW    (TM * SA_STRIDE)        // 33024 dwords
#define WBUF_DW  (D_ * SW_STRIDE)        // 18432 dwords per buffer
#define LDS_BYTES ((SA_DW + 2 * WBUF_DW) * 4)   // 279,552 B <= 320KB/WGP

// ---- async copy-engine helpers (ASYNCcnt-tracked, LDS written directly) ----
__device__ __forceinline__ void async_b128(unsigned lds_byte, const float* g) {
  asm volatile("global_load_async_to_lds_b128 %0, %1, off"
               :: "v"(lds_byte), "v"(g) : "memory");
}

__device__ __forceinline__ void wait_async0() {
#if defined(__has_builtin)
#if __has_builtin(__builtin_amdgcn_s_wait_asynccnt)
  __builtin_amdgcn_s_wait_asynccnt(0);
  return;
#else
  asm volatile("s_wait_asynccnt 0x0" ::: "memory");
  return;
#endif
#else
  asm volatile("s_wait_asynccnt 0x0" ::: "memory");
#endif
}

// stage a 512-row x KC-float W chunk: src row n at gsrc + n*gstride (already
// column-offset), dst LDS dword base lds_dw (SW_STRIDE-padded rows)
__device__ __forceinline__ void stage_w_async(const float* __restrict__ gsrc,
                                              int gstride, unsigned lds_dw,
                                              int tid) {
#pragma unroll 4
  for (int v = 0; v < 16; ++v) {
    const int i4 = v * 256 + tid;        // float4 index 0..4095
    const int n  = i4 >> 3;              // 8 float4 per row
    const int kq = i4 & 7;
    async_b128((lds_dw + (unsigned)(n * SW_STRIDE + kq * 4)) * 4u,
               gsrc + (size_t)n * gstride + kq * 4);
  }
}

// stage TM x 512 fp32 A rows into SA_STRIDE-padded LDS
__device__ __forceinline__ void stage_a_async(const float* __restrict__ gsrc,
                                              int tid) {
#pragma unroll 4
  for (int v = 0; v < 32; ++v) {
    const int i4 = v * 256 + tid;        // float4 index 0..8191
    const int r  = i4 >> 7;              // 128 float4 per row
    const int kq = i4 & 127;
    async_b128((unsigned)(r * SA_STRIDE + kq * 4) * 4u,
               gsrc + (size_t)r * D_ + kq * 4);
  }
}

// ---- GEMM inner chunk: acc += shA(64 x KC @ kc) * shW(512 x KC)^T ----------
__device__ __forceinline__ void gemm_chunk(const float* __restrict__ sh_a,
                                           const float* __restrict__ sh_w,
                                           v8f acc[4][4],
                                           int kc, int w, int lane) {
  const int mrow = lane & 15;          // M (for A) / N (for B) within tile
  const int c2   = (lane >> 4) << 1;   // upper half-wave holds K+2,K+3
#pragma unroll
  for (int kk = 0; kk < KC; kk += 4) {
    v2f af[4], bf[4];
#pragma unroll
    for (int mi = 0; mi < 4; ++mi)
      af[mi] = *(const v2f*)&sh_a[(mi * 16 + mrow) * SA_STRIDE + kc + kk + c2];
#pragma unroll
    for (int ni = 0; ni < 4; ++ni)
      bf[ni] = *(const v2f*)&sh_w[((w * 4 + ni) * 16 + mrow) * SW_STRIDE + kk + c2];
#pragma unroll
    for (int mi = 0; mi < 4; ++mi)
#pragma unroll
      for (int ni = 0; ni < 4; ++ni)
        acc[mi][ni] = __builtin_amdgcn_wmma_f32_16x16x4_f32(
            false, af[mi], false, bf[ni], (short)0, acc[mi][ni], false, false);
  }
}

__device__ __forceinline__ void zero_acc(v8f acc[4][4]) {
#pragma unroll
  for (int mi = 0; mi < 4; ++mi)
#pragma unroll
    for (int ni = 0; ni < 4; ++ni)
#pragma unroll
      for (int e = 0; e < 8; ++e) acc[mi][ni][e] = 0.0f;
}

// ---------------------------------------------------------------------------
// Kernel 1: ws[half][m][d] = sum_k pf[m][k] * W1[d][half*512 + k]
// grid = (M1/TM, 2), block = 256 (8 waves)
// ---------------------------------------------------------------------------
__global__ void __launch_bounds__(256)
p2r_layer1(const float* __restrict__ pf, const float* __restrict__ W1,
           float* __restrict__ ws) {
  extern __shared__ float smem[];
  float* sh_a = smem;                    // [TM][SA_STRIDE]

  const int tid     = threadIdx.x;
  const int half    = blockIdx.y;
  const int koff    = half * D_;
  const int rowBase = blockIdx.x * TM;
  const float* wsrc = W1 + koff;         // row stride 2*D_
  float* dst = ws + (size_t)half * M1 * D_;

  // async: A rows + W chunk 0 into buffer 0
  stage_a_async(pf + (size_t)rowBase * D_, tid);
  stage_w_async(wsrc, 2 * D_, SA_DW, tid);
  wait_async0();
  __syncthreads();

  const int w    = tid >> 5;
  const int lane = tid & 31;
  v8f acc[4][4];
  zero_acc(acc);

  for (int c = 0; c < NCHUNK; ++c) {
    if (c + 1 < NCHUNK)                  // prefetch next chunk into other buffer
      stage_w_async(wsrc + (c + 1) * KC, 2 * D_,
                    SA_DW + (unsigned)(((c + 1) & 1) * WBUF_DW), tid);
    gemm_chunk(sh_a, smem + SA_DW + (c & 1) * WBUF_DW, acc, c * KC, w, lane);
    wait_async0();
    __syncthreads();
  }

  const int mrow = lane & 15;
  const int mup  = (lane >> 4) << 3;     // upper half-wave holds M+8..M+15
#pragma unroll
  for (int mi = 0; mi < 4; ++mi)
#pragma unroll
    for (int ni = 0; ni < 4; ++ni) {
      const int col = (w * 4 + ni) * 16 + mrow;
#pragma unroll
      for (int r = 0; r < 8; ++r) {
        const int m = rowBase + mi * 16 + mup + r;
        dst[(size_t)m * D_ + col] = acc[mi][ni][r];
      }
    }
}

// ---------------------------------------------------------------------------
// Kernel 2: decode pair rows, fused bias+relu into LDS, GEMM vs W2, +b2
// grid = (M2/TM, 1), block = 256
// ---------------------------------------------------------------------------
__global__ void __launch_bounds__(256)
p2r_layer2(const float* __restrict__ ws, const float* __restrict__ b1,
           const float* __restrict__ W2, const float* __restrict__ b2,
           float* __restrict__ out) {
  extern __shared__ float smem[];
  float* sh_h = smem;                    // [TM][SA_STRIDE]
  const float* Aws = ws;
  const float* Bws = ws + (size_t)M1 * D_;

  const int tid     = threadIdx.x;
  const int rowBase = blockIdx.x * TM;   // pair-row base

  // kick off W2 chunk 0 copy; overlaps with the h build below
  stage_w_async(W2, D_, SA_DW, tid);

  // phase 1: h = relu(Aws[b,i] + Bws[b,j] + b1) staged directly in LDS
#pragma unroll 4
  for (int v = 0; v < 32; ++v) {
    const int i4 = v * 256 + tid;
    const int r  = i4 >> 7;
    const int kq = i4 & 127;
    const unsigned g = (unsigned)(rowBase + r);
    const unsigned b = g / PP;
    const unsigned p = g - b * PP;
    const unsigned i = p / 31u;
    const unsigned q = p - i * 31u;
    const unsigned j = q + (q >= i ? 1u : 0u);
    const float4 a4 = ((const float4*)(Aws + (size_t)(b * NP + i) * D_))[kq];
    const float4 c4 = ((const float4*)(Bws + (size_t)(b * NP + j) * D_))[kq];
    const float4 s4 = ((const float4*)b1)[kq];
    float4 h;
    h.x = fmaxf(a4.x + c4.x + s4.x, 0.0f);
    h.y = fmaxf(a4.y + c4.y + s4.y, 0.0f);
    h.z = fmaxf(a4.z + c4.z + s4.z, 0.0f);
    h.w = fmaxf(a4.w + c4.w + s4.w, 0.0f);
    *(float4*)&sh_h[r * SA_STRIDE + kq * 4] = h;
  }
  wait_async0();
  __syncthreads();

  const int w    = tid >> 5;
  const int lane = tid & 31;
  v8f acc[4][4];
  zero_acc(acc);

  for (int c = 0; c < NCHUNK; ++c) {
    if (c + 1 < NCHUNK)
      stage_w_async(W2 + (c + 1) * KC, D_,
                    SA_DW + (unsigned)(((c + 1) & 1) * WBUF_DW), tid);
    gemm_chunk(sh_h, smem + SA_DW + (c & 1) * WBUF_DW, acc, c * KC, w, lane);
    wait_async0();
    __syncthreads();
  }

  const int mrow = lane & 15;
  const int mup  = (lane >> 4) << 3;
#pragma unroll
  for (int mi = 0; mi < 4; ++mi)
#pragma unroll
    for (int ni = 0; ni < 4; ++ni) {
      const int col  = (w * 4 + ni) * 16 + mrow;
      const float bv = b2[col];
#pragma unroll
      for (int r = 0; r < 8; ++r) {
        const int m = rowBase + mi * 16 + mup + r;
        out[(size_t)m * D_ + col] = acc[mi][ni][r] + bv;
      }
    }
}

// ---------------------------------------------------------------------------
extern "C" void kernel_launch(void* const* d_in, const int* in_sizes, int n_in,
                              void* d_out, int out_size, void* d_ws,
                              size_t ws_size, hipStream_t stream) {
  (void)in_sizes; (void)n_in; (void)out_size; (void)ws_size;
  const float* pf = (const float*)d_in[0];
  const float* W1 = (const float*)d_in[1];
  const float* b1 = (const float*)d_in[2];
  const float* W2 = (const float*)d_in[3];
  const float* b2 = (const float*)d_in[4];
  float* out = (float*)d_out;
  float* ws  = (float*)d_ws;   // needs 2 * M1 * D_ * 4 = 8 MB

  (void)hipFuncSetAttribute((const void*)p2r_layer1,
                            hipFuncAttributeMaxDynamicSharedMemorySize, LDS_BYTES);
  (void)hipFuncSetAttribute((const void*)p2r_layer2,
                            hipFuncAttributeMaxDynamicSharedMemorySize, LDS_BYTES);

  dim3 blk(256, 1, 1);
  dim3 g1(M1 / TM, 2, 1);
  hipLaunchKernelGGL(p2r_layer1, g1, blk, LDS_BYTES, stream, pf, W1, ws);
  dim3 g2(M2 / TM, 1, 1);
  hipLaunchKernelGGL(p2r_layer2, g2, blk, LDS_BYTES, stream, ws, b1, W2, b2, out);
}